// HashEncoder_40681930227963
// MI455X (gfx1250) — compile-verified
//
#include <hip/hip_runtime.h>
#include <stdint.h>

typedef float v2f __attribute__((ext_vector_type(2)));
typedef float v4f __attribute__((ext_vector_type(4)));

namespace {

constexpr int  NLEV    = 16;
constexpr uint32_t PRIME_Y = 2654435761u;
constexpr uint32_t PRIME_Z = 805459861u;

// Exact replica of HashEncoder.__init__ offset logic:
//   res = ceil(16 * 2^i); p = min(2^19, (res+1)^3); p = (p/8)*8
constexpr uint32_t level_params(int i) {
  unsigned long long res = 16ull << i;
  unsigned long long p   = (res + 1ull) * (res + 1ull) * (res + 1ull);
  constexpr unsigned long long maxp = 1ull << 19;
  if (p > maxp) p = maxp;
  p = (p / 8ull) * 8ull;
  return (uint32_t)p;
}
constexpr uint32_t level_offset(int i) {
  uint32_t off = 0;
  for (int k = 0; k < i; ++k) off += level_params(k);
  return off;
}

} // namespace

__global__ __launch_bounds__(256)
void hashenc_fwd(const float* __restrict__ xyz,
                 const float* __restrict__ emb,
                 float* __restrict__ out,
                 int B)
{
  const int b = (int)(blockIdx.x * 256u + threadIdx.x);
  if (b >= B) return;

  // xyz is streamed exactly once -> non-temporal loads (keep L2 for the table)
  const float* xp = xyz + (long long)b * 3;
  const float x = __builtin_nontemporal_load(xp + 0);
  const float y = __builtin_nontemporal_load(xp + 1);
  const float z = __builtin_nontemporal_load(xp + 2);

  float res[2 * NLEV];

  // Full unroll: per-level hashmap sizes become literal constants, so the
  // non-power-of-two moduli (levels 0..2) fold to magic-multiply sequences,
  // and the scheduler can keep many independent b64 gathers in flight.
  #pragma unroll
  for (int l = 0; l < NLEV; ++l) {
    const uint32_t hs   = level_params(l);          // compile-time constant
    const uint32_t off  = level_offset(l);          // compile-time constant
    const bool     pow2 = (hs & (hs - 1u)) == 0u;   // levels >= 3
    const float scale   = (float)(16u << l) - 1.0f; // exp2(l)*16 - 1, exact in fp32

    const float px = fmaf(x, scale, 0.5f);
    const float py = fmaf(y, scale, 0.5f);
    const float pz = fmaf(z, scale, 0.5f);
    const float gx = floorf(px), gy = floorf(py), gz = floorf(pz);
    const float fx = px - gx,    fy = py - gy,    fz = pz - gz;
    const uint32_t ix = (uint32_t)gx;
    const uint32_t iy = (uint32_t)gy;
    const uint32_t iz = (uint32_t)gz;

    // torch-ngp fast_hash: primes {1, 2654435761, 805459861}, xor-combined.
    // (Every level takes the hash path: (res+1)^3 is odd, so rounding to a
    //  multiple of 8 always makes the table smaller than the dense grid.)
    const uint32_t hx0 = ix;                 const uint32_t hx1 = ix + 1u;
    const uint32_t hy0 = iy * PRIME_Y;       const uint32_t hy1 = (iy + 1u) * PRIME_Y;
    const uint32_t hz0 = iz * PRIME_Z;       const uint32_t hz1 = (iz + 1u) * PRIME_Z;

    uint32_t i000 = hx0 ^ hy0 ^ hz0;
    uint32_t i100 = hx1 ^ hy0 ^ hz0;
    uint32_t i010 = hx0 ^ hy1 ^ hz0;
    uint32_t i110 = hx1 ^ hy1 ^ hz0;
    uint32_t i001 = hx0 ^ hy0 ^ hz1;
    uint32_t i101 = hx1 ^ hy0 ^ hz1;
    uint32_t i011 = hx0 ^ hy1 ^ hz1;
    uint32_t i111 = hx1 ^ hy1 ^ hz1;

    if (pow2) {
      const uint32_t m = hs - 1u;
      i000 &= m; i100 &= m; i010 &= m; i110 &= m;
      i001 &= m; i101 &= m; i011 &= m; i111 &= m;
    } else {
      i000 %= hs; i100 %= hs; i010 %= hs; i110 %= hs;
      i001 %= hs; i101 %= hs; i011 %= hs; i111 %= hs;
    }

    // Issue all 8 gathers (global_load_b64) before consuming any of them.
    const v2f* __restrict__ tab = (const v2f*)emb + off;
    const v2f e000 = tab[i000];
    const v2f e100 = tab[i100];
    const v2f e010 = tab[i010];
    const v2f e110 = tab[i110];
    const v2f e001 = tab[i001];
    const v2f e101 = tab[i101];
    const v2f e011 = tab[i011];
    const v2f e111 = tab[i111];

    const float wx0 = 1.0f - fx, wx1 = fx;
    const float wy0 = 1.0f - fy, wy1 = fy;
    const float wz0 = 1.0f - fz, wz1 = fz;

    const float w000 = wx0 * wy0 * wz0;
    const float w100 = wx1 * wy0 * wz0;
    const float w010 = wx0 * wy1 * wz0;
    const float w110 = wx1 * wy1 * wz0;
    const float w001 = wx0 * wy0 * wz1;
    const float w101 = wx1 * wy0 * wz1;
    const float w011 = wx0 * wy1 * wz1;
    const float w111 = wx1 * wy1 * wz1;

    // Accumulate in the reference's corner order c = 0..7
    float a0 = 0.0f, a1 = 0.0f;
    a0 = fmaf(w000, e000.x, a0);  a1 = fmaf(w000, e000.y, a1);
    a0 = fmaf(w100, e100.x, a0);  a1 = fmaf(w100, e100.y, a1);
    a0 = fmaf(w010, e010.x, a0);  a1 = fmaf(w010, e010.y, a1);
    a0 = fmaf(w110, e110.x, a0);  a1 = fmaf(w110, e110.y, a1);
    a0 = fmaf(w001, e001.x, a0);  a1 = fmaf(w001, e001.y, a1);
    a0 = fmaf(w101, e101.x, a0);  a1 = fmaf(w101, e101.y, a1);
    a0 = fmaf(w011, e011.x, a0);  a1 = fmaf(w011, e011.y, a1);
    a0 = fmaf(w111, e111.x, a0);  a1 = fmaf(w111, e111.y, a1);

    res[2 * l + 0] = a0;
    res[2 * l + 1] = a1;
  }

  // One point owns one full 128B line of output: 8 x b128 non-temporal stores
  // so the 128MB output stream doesn't evict the 57MB table from the 192MB L2.
  v4f* op = (v4f*)(out + (long long)b * (2 * NLEV));
  #pragma unroll
  for (int i = 0; i < 2 * NLEV / 4; ++i) {
    v4f v = { res[4 * i + 0], res[4 * i + 1], res[4 * i + 2], res[4 * i + 3] };
    __builtin_nontemporal_store(v, op + i);
  }
}

extern "C" void kernel_launch(void* const* d_in, const int* in_sizes, int n_in,
                              void* d_out, int out_size, void* d_ws, size_t ws_size,
                              hipStream_t stream) {
  const float* xyz = (const float*)d_in[0];
  const float* emb = (const float*)d_in[1];
  // d_in[2] is `normalize` == 0 -> raw-coordinate path (what we implement)
  float* out = (float*)d_out;

  const int B = in_sizes[0] / 3;
  const int threads = 256;                 // 8 wave32 waves per block
  const int blocks  = (B + threads - 1) / threads;
  hipLaunchKernelGGL(hashenc_fwd, dim3(blocks), dim3(threads), 0, stream,
                     xyz, emb, out, B);
}